// AttentionORLiTLayer_19610820673552
// MI455X (gfx1250) — compile-verified
//
#include <hip/hip_runtime.h>
#include <hip/hip_bf16.h>
#include <math.h>
#include <stdint.h>

#define SEQ 4096
#define INPUT_DIM 512
#define HEAD_NUM 8
#define HEAD_DIM 64
#define ETA 4
#define RR 8
#define EHD (ETA * HEAD_DIM)            // 256
#define NS (HEAD_NUM * EHD)             // 2048
#define NV (HEAD_NUM * HEAD_DIM)        // 512
#define EPSV 1e-5f

#define CHUNK 32
#define NCHUNK 128                      // CHUNK*NCHUNK == SEQ
#define NKEY (RR * NS)                  // 16384
#define NVAL (RR * NV)                  // 4096
#define NSTATE (NKEY + NS + NVAL)       // 22528
#define OFF_S NKEY
#define OFF_V (NKEY + NS)

#define KC 16                           // GEMM K-chunk staged in LDS

typedef float v2f __attribute__((ext_vector_type(2)));
typedef float v8f __attribute__((ext_vector_type(8)));
typedef unsigned int u32x4 __attribute__((ext_vector_type(4)));
typedef int i32x4 __attribute__((ext_vector_type(4)));
typedef int i32x8 __attribute__((ext_vector_type(8)));

#if __has_builtin(__builtin_amdgcn_tensor_load_to_lds)
#define HAVE_TDM 1
#else
#define HAVE_TDM 0
#endif

#if HAVE_TDM
// ---------------------------------------------------------------------------
// Issue one TDM 2D tile load: global (row-major, row_stride elems) -> LDS.
// D# packing per CDNA5 ISA (data_size=4B, 2D tile,
// no padding / iterate / multicast). OOB columns/rows are zero-filled by HW.
// Tile is written compactly to LDS (tile_dim0 elems/row, rows contiguous).
// ---------------------------------------------------------------------------
__device__ __forceinline__ void tdm_load_2d(unsigned lds_off, const float* gptr,
                                            unsigned tile_cols, unsigned tile_rows,
                                            unsigned tensor_cols, unsigned tensor_rows,
                                            unsigned long long row_stride) {
  unsigned long long ga = (unsigned long long)(uintptr_t)gptr;
  u32x4 g0;
  g0[0] = 1u;                                            // count=1 (valid user D#)
  g0[1] = lds_off;                                       // lds_addr (bytes)
  g0[2] = (unsigned)(ga & 0xFFFFFFFFu);                  // global_addr[31:0]
  g0[3] = (unsigned)((ga >> 32) & 0x01FFFFFFu)           // global_addr[56:32]
          | (2u << 30);                                  // type=2 ("image")
  i32x8 g1;
  g1[0] = (int)(2u << 16);                               // data_size=2 -> 4 bytes
  g1[1] = (int)((tensor_cols & 0xFFFFu) << 16);          // tensor_dim0[15:0]
  g1[2] = (int)(((tensor_cols >> 16) & 0xFFFFu)          // tensor_dim0[31:16]
          | ((tensor_rows & 0xFFFFu) << 16));            // tensor_dim1[15:0]
  g1[3] = (int)(((tensor_rows >> 16) & 0xFFFFu)          // tensor_dim1[31:16]
          | ((tile_cols & 0xFFFFu) << 16));              // tile_dim0
  g1[4] = (int)(tile_rows & 0xFFFFu);                    // tile_dim1 (tile_dim2=0)
  g1[5] = (int)(row_stride & 0xFFFFFFFFull);             // tensor_dim0_stride[31:0]
  g1[6] = (int)((row_stride >> 32) & 0xFFFFull);         // stride[47:32]; dim1_stride=0
  g1[7] = 0;
  i32x4 z4 = {0, 0, 0, 0};
#if defined(__clang_major__) && __clang_major__ >= 23
  i32x8 z8 = {0, 0, 0, 0, 0, 0, 0, 0};
  __builtin_amdgcn_tensor_load_to_lds(g0, g1, z4, z4, z8, 0);
#else
  __builtin_amdgcn_tensor_load_to_lds(g0, g1, z4, z4, 0);
#endif
}
#endif

// ---------------------------------------------------------------------------
// FP32 WMMA GEMM: C[M,N] = A[M,K] * B[K,N] + bias (all row-major).
// block = 128 threads (4 waves) -> 16x256 tile; each wave owns a 16x64 slice
// with PRIVATE LDS staging (own A copy + own B slice), TDM double-buffered per
// wave: issue chunk k+1, s_wait_tensorcnt 2 (in-order => chunk k resident),
// compute. Zero workgroup barriers; branch-free WMMA inner loop.
// ---------------------------------------------------------------------------
__global__ __launch_bounds__(128)
void wmma_gemm_f32(const float* __restrict__ A, const float* __restrict__ B,
                   const float* __restrict__ bias, float* __restrict__ C,
                   int M, int N, int K) {
  __shared__ float lA[2][4][16 * KC];   // [buf][wave][mRow*KC + k]   8 KB
  __shared__ float lB[2][4][KC * 64];   // [buf][wave][krow*64 + col] 32 KB
  const int tid = threadIdx.x;
  const int wave = __builtin_amdgcn_readfirstlane(tid >> 5);
  const int lane = tid & 31;
  const int mBase = blockIdx.x * 16;
  const int nBlock = blockIdx.y * 256;
  const int nWave = nBlock + wave * 64;       // this wave's 64-col slice
  if (nWave >= N) return;                     // wave-uniform exit
  const int half = lane >> 4;                 // 0/1
  const int mRow = lane & 15;
  const int nCol = lane & 15;
  const int kb = half * 2;                    // fragment K sub-offset
  const unsigned nRem = (unsigned)(N - nWave);

  v8f acc[4] = {v8f{}, v8f{}, v8f{}, v8f{}};
  const int nKC = K / KC;

#if HAVE_TDM
  tdm_load_2d((unsigned)(uintptr_t)&lA[0][wave][0], A + (size_t)mBase * K,
              KC, 16, (unsigned)K, (unsigned)(M - mBase), (unsigned long long)K);
  tdm_load_2d((unsigned)(uintptr_t)&lB[0][wave][0], B + nWave,
              64, KC, nRem, (unsigned)K, (unsigned long long)N);
#endif
  for (int kc = 0; kc < nKC; ++kc) {
    const int buf = kc & 1;
    const int k0 = kc * KC;
#if HAVE_TDM
    if (kc + 1 < nKC) {                       // software pipeline: issue next
      const int k1 = k0 + KC;
      tdm_load_2d((unsigned)(uintptr_t)&lA[buf ^ 1][wave][0],
                  A + (size_t)mBase * K + k1, KC, 16,
                  (unsigned)(K - k1), (unsigned)(M - mBase), (unsigned long long)K);
      tdm_load_2d((unsigned)(uintptr_t)&lB[buf ^ 1][wave][0],
                  B + (size_t)k1 * N + nWave, 64, KC,
                  nRem, (unsigned)(K - k1), (unsigned long long)N);
      __builtin_amdgcn_s_wait_tensorcnt(2);   // chunk kc resident (in-order)
    } else {
      __builtin_amdgcn_s_wait_tensorcnt(0);
    }
#else
    // Fallback: per-wave synchronous staging (same-wave LDS ops are in-order,
    // so no barrier is needed).
    for (int i = lane; i < 16 * KC; i += 32)
      lA[buf][wave][i] = A[(size_t)(mBase + i / KC) * K + k0 + (i % KC)];
    for (int i = lane; i < KC * 64; i += 32) {
      int r = i >> 6, cl = i & 63;
      lB[buf][wave][i] =
          ((unsigned)cl < nRem) ? B[(size_t)(k0 + r) * N + nWave + cl] : 0.f;
    }
#endif
    // Branch-free WMMA inner loop from this wave's private LDS slices.
#pragma unroll
    for (int kk = 0; kk < KC; kk += 4) {
      v2f a = *(const v2f*)&lA[buf][wave][mRow * KC + kk + kb];
#pragma unroll
      for (int t = 0; t < 4; ++t) {
        v2f b;
        b.x = lB[buf][wave][(kk + kb) * 64 + t * 16 + nCol];
        b.y = lB[buf][wave][(kk + kb + 1) * 64 + t * 16 + nCol];
        acc[t] = __builtin_amdgcn_wmma_f32_16x16x4_f32(
            false, a, false, b, (short)0, acc[t], false, false);
      }
    }
  }
#pragma unroll
  for (int t = 0; t < 4; ++t) {
    int n = nWave + t * 16 + nCol;
    if (n < N) {
      float bv = bias[n];
#pragma unroll
      for (int v = 0; v < 8; ++v) {
        int row = mBase + half * 8 + v;       // C/D layout: VGPR v -> M=v (+8)
        C[(size_t)row * N + n] = acc[t][v] + bv;
      }
    }
  }
}

// ---------------------------------------------------------------------------
// Elementwise epilogue: gates, outer products, discounts, oscillators.
// ---------------------------------------------------------------------------
__global__ void epilogue_kernel(const float* __restrict__ kqv, const float* __restrict__ p,
                                const float* __restrict__ term, const int* __restrict__ tick,
                                float* __restrict__ keysS, float* __restrict__ dgam,
                                float* __restrict__ qout, float* __restrict__ vb,
                                float* __restrict__ dbeta, float* __restrict__ occil) {
  int idx = blockIdx.x * blockDim.x + threadIdx.x;
  if (idx >= SEQ * HEAD_NUM * HEAD_DIM) return;
  const int d = idx & 63;
  const int h = (idx >> 6) & 7;
  const int c = idx >> 9;
  const float* row = kqv + (size_t)c * (HEAD_NUM * 5 * HEAD_DIM) + h * (5 * HEAD_DIM);
  float kk = row[d];
  float qq = row[HEAD_DIM + d];
  float vv = row[2 * HEAD_DIM + d];
  float be = row[3 * HEAD_DIM + d];
  float gg = row[4 * HEAD_DIM + d];
  const float* prow = p + (size_t)c * (HEAD_NUM * 3 * ETA) + h * (3 * ETA);
  float nt = 1.f - term[c];
  float rk = fmaxf(kk, 0.f);
  float rq = fmaxf(qq, 0.f);
  float sg = 1.f / (1.f + __expf(-gg));
  float sb = 1.f / (1.f + __expf(-be));
#pragma unroll
  for (int n = 0; n < ETA; ++n) {
    float p1 = fmaxf(prow[n], 0.f);
    float p2 = fmaxf(prow[ETA + n], 0.f);
    float p3 = 1.f / (1.f + __expf(-prow[2 * ETA + n]));
    float gam = sg * p3;
    size_t o = (size_t)c * NS + h * EHD + n * HEAD_DIM + d;
    keysS[o] = rk * p1 * gam;             // s = keys * gammas
    dgam[o] = (1.f - gam) * nt;           // discount_gamma
    qout[o] = rq * p2;                    // queries
  }
  size_t ov = (size_t)c * NV + h * HEAD_DIM + d;
  vb[ov] = vv * sb;                       // values * beta
  dbeta[ov] = (1.f - sb) * nt;            // discount_beta
  if (h == 0 && d < RR) {
    const float PI = 3.14159265358979323846f;
    float omega = -PI + (2.f * PI) * (float)d / (float)(RR - 1);
    occil[(size_t)c * RR + d] = __cosf((float)(tick[0] + c + 1) * omega);
  }
}

// ---------------------------------------------------------------------------
// Scan pass A: per-chunk (A = prod discounts, B = chunk-local discounted sum).
// grid = (NCHUNK, 10): roles 0..7 key+s chains, 8..9 value chains.
// ---------------------------------------------------------------------------
__global__ void scan_passA(const float* __restrict__ keysS, const float* __restrict__ dgam,
                           const float* __restrict__ vb, const float* __restrict__ dbeta,
                           const float* __restrict__ occil,
                           float* __restrict__ Ag, float* __restrict__ Bg) {
  const int g = blockIdx.x;
  const int role = blockIdx.y;
  const int tid = threadIdx.x;
  const int t0 = g * CHUNK;
  float* AgC = Ag + (size_t)g * NSTATE;
  float* BgC = Bg + (size_t)g * NSTATE;
  if (role < 8) {
    const int idx = role * 256 + tid;     // h*256 + d
    float Ak[RR], Bk[RR];
#pragma unroll
    for (int r = 0; r < RR; ++r) { Ak[r] = 1.f; Bk[r] = 0.f; }
    float As = 1.f, Bs = 0.f;
    for (int t = t0; t < t0 + CHUNK; ++t) {
      float ks = keysS[(size_t)t * NS + idx];
      float dg = dgam[(size_t)t * NS + idx];
      if (t + 8 < SEQ) {
        __builtin_prefetch(keysS + (size_t)(t + 8) * NS + idx, 0, 1);
        __builtin_prefetch(dgam + (size_t)(t + 8) * NS + idx, 0, 1);
      }
#pragma unroll
      for (int r = 0; r < RR; ++r) {
        float oc = occil[t * RR + r];
        Ak[r] *= dg;
        Bk[r] = dg * Bk[r] + ks * oc;
      }
      As *= dg;
      Bs = dg * Bs + ks;
    }
#pragma unroll
    for (int r = 0; r < RR; ++r) {
      AgC[r * NS + idx] = Ak[r];
      BgC[r * NS + idx] = Bk[r];
    }
    AgC[OFF_S + idx] = As;
    BgC[OFF_S + idx] = Bs;
  } else {
    const int idx = (role - 8) * 256 + tid;  // h*64 + j
    float Av[RR], Bv[RR];
#pragma unroll
    for (int r = 0; r < RR; ++r) { Av[r] = 1.f; Bv[r] = 0.f; }
    for (int t = t0; t < t0 + CHUNK; ++t) {
      float vv = vb[(size_t)t * NV + idx];
      float db = dbeta[(size_t)t * NV + idx];
#pragma unroll
      for (int r = 0; r < RR; ++r) {
        float oc = occil[t * RR + r];
        Av[r] *= db;
        Bv[r] = db * Bv[r] + vv * oc;
      }
    }
#pragma unroll
    for (int r = 0; r < RR; ++r) {
      AgC[OFF_V + r * NV + idx] = Av[r];
      BgC[OFF_V + r * NV + idx] = Bv[r];
    }
  }
}

// ---------------------------------------------------------------------------
// Scan combine: chunk summaries -> chunk-start states; emits final states.
// ---------------------------------------------------------------------------
__global__ void scan_combine(const float* __restrict__ Ag, const float* __restrict__ Bg,
                             const float* __restrict__ kprev, const float* __restrict__ vprev,
                             const float* __restrict__ sprev, const int* __restrict__ tick,
                             float* __restrict__ Sstart, float* __restrict__ outF) {
  int e = blockIdx.x * blockDim.x + threadIdx.x;
  if (e >= NSTATE) return;
  float S;
  if (e < OFF_S)      S = kprev[e];
  else if (e < OFF_V) S = sprev[e - OFF_S];
  else                S = vprev[e - OFF_V];
  for (int g = 0; g < NCHUNK; ++g) {
    Sstart[(size_t)g * NSTATE + e] = S;
    S = Ag[(size_t)g * NSTATE + e] * S + Bg[(size_t)g * NSTATE + e];
  }
  if (e < OFF_S)      outF[e] = S;
  else if (e < OFF_V) outF[NKEY + NVAL + (e - OFF_S)] = S;
  else                outF[NKEY + (e - OFF_V)] = S;
  if (e == 0) outF[NKEY + NVAL + NS] = (float)(tick[0] + SEQ);
}

__device__ __forceinline__ float wave_red(float x) {
#pragma unroll
  for (int s = 16; s > 0; s >>= 1) x += __shfl_xor(x, s, 32);
  return x;
}

// ---------------------------------------------------------------------------
// Scan pass C: replay chunks (state in registers, 1024 threads/block);
// per-step kdq/norm/kv via wave32 shuffle + LDS reductions; writes attn_in.
// ---------------------------------------------------------------------------
__global__ __launch_bounds__(1024)
void scan_passC(const float* __restrict__ keysS, const float* __restrict__ dgam,
                const float* __restrict__ qarr, const float* __restrict__ vb,
                const float* __restrict__ dbeta, const float* __restrict__ occil,
                const float* __restrict__ Sstart, float* __restrict__ attn_in) {
  const int g = blockIdx.x;
  const int tid = threadIdx.x;
  const int h = tid >> 7;
  const int dd = tid & 127;
  const int d0 = dd, d1 = dd + 128;
  const int j = dd & 63;
  const int rg = dd >> 6;
  const int lane = tid & 31;
  const int w4 = (tid >> 5) & 3;

  __shared__ float ls_kdq[RR][HEAD_NUM][4];
  __shared__ float ls_norm[HEAD_NUM][4];
  __shared__ float ls_kdqF[RR][HEAD_NUM];
  __shared__ float ls_normF[HEAD_NUM];
  __shared__ float ls_kv[HEAD_NUM][HEAD_DIM][2];

  const float* S0 = Sstart + (size_t)g * NSTATE;
  float fk0[RR], fk1[RR], fv[4];
#pragma unroll
  for (int r = 0; r < RR; ++r) {
    fk0[r] = S0[r * NS + h * EHD + d0];
    fk1[r] = S0[r * NS + h * EHD + d1];
  }
  float fs0 = S0[OFF_S + h * EHD + d0];
  float fs1 = S0[OFF_S + h * EHD + d1];
#pragma unroll
  for (int i = 0; i < 4; ++i)
    fv[i] = S0[OFF_V + (rg * 4 + i) * NV + h * HEAD_DIM + j];

  for (int t = 0; t < CHUNK; ++t) {
    const int tG = g * CHUNK + t;
    const size_t bK = (size_t)tG * NS + h * EHD;
    float ks0 = keysS[bK + d0], ks1 = keysS[bK + d1];
    float dg0 = dgam[bK + d0], dg1 = dgam[bK + d1];
    float q0 = qarr[bK + d0], q1 = qarr[bK + d1];
    float oc[RR];
#pragma unroll
    for (int r = 0; r < RR; ++r) oc[r] = occil[tG * RR + r];

    fs0 = dg0 * fs0 + ks0;
    fs1 = dg1 * fs1 + ks1;
    float np = fs0 * q0 + fs1 * q1;
#pragma unroll
    for (int r = 0; r < RR; ++r) {
      fk0[r] = dg0 * fk0[r] + ks0 * oc[r];
      fk1[r] = dg1 * fk1[r] + ks1 * oc[r];
      float pp = fk0[r] * q0 + fk1[r] * q1;
      float s = wave_red(pp);
      if (lane == 0) ls_kdq[r][h][w4] = s;
    }
    {
      float s = wave_red(np);
      if (lane == 0) ls_norm[h][w4] = s;
    }
    __syncthreads();
    if (tid < RR * HEAD_NUM) {
      int r = tid >> 3, hh = tid & 7;
      ls_kdqF[r][hh] =
          ls_kdq[r][hh][0] + ls_kdq[r][hh][1] + ls_kdq[r][hh][2] + ls_kdq[r][hh][3];
    }
    if (tid < HEAD_NUM)
      ls_normF[tid] =
          ls_norm[tid][0] + ls_norm[tid][1] + ls_norm[tid][2] + ls_norm[tid][3];
    __syncthreads();
    const size_t bV = (size_t)tG * NV + h * HEAD_DIM + j;
    float vv = vb[bV], db = dbeta[bV];
    float kvp = 0.f;
#pragma unroll
    for (int i = 0; i < 4; ++i) {
      int r = rg * 4 + i;
      fv[i] = db * fv[i] + vv * oc[r];
      kvp += fv[i] * ls_kdqF[r][h];
    }
    ls_kv[h][j][rg] = kvp;
    __syncthreads();
    if (rg == 0) {
      float kv = ls_kv[h][j][0] + ls_kv[h][j][1];
      attn_in[(size_t)tG * NV + h * HEAD_DIM + j] =
          kv / (2.f * (float)RR * ls_normF[h] + EPSV);
    }
  }
}

// ---------------------------------------------------------------------------
extern "C" void kernel_launch(void* const* d_in, const int* in_sizes, int n_in,
                              void* d_out, int out_size, void* d_ws, size_t ws_size,
                              hipStream_t stream) {
  const float* inputs = (const float*)d_in[0];
  const float* term   = (const float*)d_in[1];
  const float* kprev  = (const float*)d_in[2];
  const float* vprev  = (const float*)d_in[3];
  const float* sprev  = (const float*)d_in[4];
  const int*   tick   = (const int*)d_in[5];
  const float* Wkqv   = (const float*)d_in[6];
  const float* bkqv   = (const float*)d_in[7];
  const float* Wp     = (const float*)d_in[8];
  const float* bp     = (const float*)d_in[9];
  const float* Wproj  = (const float*)d_in[10];
  const float* bproj  = (const float*)d_in[11];
  float* out  = (float*)d_out;
  float* outF = out + (size_t)SEQ * INPUT_DIM;

  float* ws = (float*)d_ws;
  size_t o = 0;
  float* kqv   = ws + o;  o += (size_t)SEQ * (HEAD_NUM * 5 * HEAD_DIM);
  float* pbuf  = ws + o;  o += (size_t)SEQ * (HEAD_NUM * 3 * ETA);
  float* keysS = ws + o;  o += (size_t)SEQ * NS;
  float* dgam  = ws + o;  o += (size_t)SEQ * NS;
  float* qbuf  = ws + o;  o += (size_t)SEQ * NS;
  float* vbb   = ws + o;  o += (size_t)SEQ * NV;
  float* dbeta = ws + o;  o += (size_t)SEQ * NV;
  float* occil = ws + o;  o += (size_t)SEQ * RR;
  float* Ag    = ws + o;  o += (size_t)NCHUNK * NSTATE;
  float* Bg    = ws + o;  o += (size_t)NCHUNK * NSTATE;
  float* Sst   = ws + o;  o += (size_t)NCHUNK * NSTATE;
  float* attn  = ws + o;  o += (size_t)SEQ * INPUT_DIM;

  wmma_gemm_f32<<<dim3(SEQ / 16, (HEAD_NUM * 5 * HEAD_DIM + 255) / 256), 128, 0, stream>>>(
      inputs, Wkqv, bkqv, kqv, SEQ, HEAD_NUM * 5 * HEAD_DIM, INPUT_DIM);
  wmma_gemm_f32<<<dim3(SEQ / 16, 1), 128, 0, stream>>>(
      inputs, Wp, bp, pbuf, SEQ, HEAD_NUM * 3 * ETA, INPUT_DIM);
  epilogue_kernel<<<(SEQ * HEAD_NUM * HEAD_DIM) / 256, 256, 0, stream>>>(
      kqv, pbuf, term, tick, keysS, dgam, qbuf, vbb, dbeta, occil);
  scan_passA<<<dim3(NCHUNK, 10), 256, 0, stream>>>(keysS, dgam, vbb, dbeta, occil, Ag, Bg);
  scan_combine<<<(NSTATE + 255) / 256, 256, 0, stream>>>(
      Ag, Bg, kprev, vprev, sprev, tick, Sst, outF);
  scan_passC<<<NCHUNK, 1024, 0, stream>>>(keysS, dgam, qbuf, vbb, dbeta, occil, Sst, attn);
  wmma_gemm_f32<<<dim3(SEQ / 16, INPUT_DIM / 256), 128, 0, stream>>>(
      attn, Wproj, bproj, out, SEQ, INPUT_DIM, INPUT_DIM);
}